// Fractal1D_50912542327443
// MI455X (gfx1250) — compile-verified
//
#include <hip/hip_runtime.h>
#include <hip/hip_bf16.h>
#include <math.h>

typedef __attribute__((ext_vector_type(2))) float v2f;
typedef __attribute__((ext_vector_type(8))) float v8f;

#define NUMV 3
#define NVD 6
#define MAXD_CAP 12
#define NWAVES 8          // 256 threads / wave32

__device__ __forceinline__ float sigm(float z) { return 1.0f / (1.0f + __expf(-z)); }
__device__ __forceinline__ int   rfli(int v)   { return __builtin_amdgcn_readfirstlane(v); }
__device__ __forceinline__ float rflf(float v) {
    return __uint_as_float(__builtin_amdgcn_readfirstlane(__float_as_uint(v)));
}

__global__ __launch_bounds__(256)
void Fractal1D_kernel(const float* __restrict__ x,
                      const float* __restrict__ spp,
                      const float* __restrict__ mats,           // [2,6,6]
                      const unsigned char* __restrict__ lmask,  // [6,6] bool
                      const unsigned char* __restrict__ rmask,  // [6,6] bool
                      const int* __restrict__ maxd_p,
                      float* __restrict__ out,                  // [npts,3]
                      int npts)
{
    // ---- x-independent tree tables ----
    __shared__ float Lsplit[NVD];
    __shared__ float Lp[NVD * 4];     // per node idx: [pl0, pl1, pr0, pr1]
    __shared__ int   Lcid[NVD * 4];   // child node idx
    __shared__ int   Lcls[NVD * 4];   // child class (idx % 3)

    // ---- per-wave DFS stack in LDS (kills VGPR waterfall indexing) ----
    __shared__ int   Sid[NWAVES][MAXD_CAP];        // uniform
    __shared__ float Sst[NWAVES][MAXD_CAP];        // uniform
    __shared__ float Sen[NWAVES][MAXD_CAP];        // uniform
    __shared__ int   Scc[NWAVES][MAXD_CAP];        // uniform child counter
    __shared__ float Sw [NWAVES][MAXD_CAP][32];    // per-lane weight
    __shared__ float St [NWAVES][MAXD_CAP][32];    // per-lane sigmoid t

    if (threadIdx.x == 0) {
        for (int i = 0; i < NVD; ++i) {
            float p = fminf(fmaxf(spp[i], 0.0f), 1.0f);
            Lsplit[i] = sigm(p * 4.0f - 2.0f);
        }
        for (int row = 0; row < NVD; ++row) {
            for (int side = 0; side < 2; ++side) {
                const float* M = mats + side * NVD * NVD + row * NVD;
                const unsigned char* Mk = (side == 0 ? lmask : rmask) + row * NVD;
                int j1 = 0; float v1 = -3.0e10f;
                for (int j = 0; j < NVD; ++j) {
                    float v = Mk[j] ? M[j] : -1.0e10f;
                    if (v > v1) { v1 = v; j1 = j; }
                }
                int j2 = 0; float v2 = -3.0e10f;
                for (int j = 0; j < NVD; ++j) {
                    if (j == j1) continue;
                    float v = Mk[j] ? M[j] : -1.0e10f;
                    if (v > v2) { v2 = v; j2 = j; }
                }
                float r  = __expf((v2 - v1) * 10.0f);   // stable 2-way softmax*10
                float p1 = 1.0f / (1.0f + r);
                float p2 = r / (1.0f + r);
                if (p1 <= 1e-5f) p1 = 0.0f;
                if (p2 <= 1e-5f) p2 = 0.0f;
                int o = row * 4 + side * 2;
                Lp[o + 0] = p1;          Lp[o + 1] = p2;
                Lcid[o + 0] = j1;        Lcid[o + 1] = j2;
                Lcls[o + 0] = j1 % NUMV; Lcls[o + 1] = j2 % NUMV;
            }
        }
    }
    __syncthreads();

    const int maxd  = *maxd_p;
    const int lane  = threadIdx.x & 31;
    const int wib   = threadIdx.x >> 5;        // wave in block
    const int wave  = (blockIdx.x * blockDim.x + threadIdx.x) >> 5;
    const int tileBase = wave * 16;            // 16 points per wave
    const int laneN = lane & 15;
    const bool hi   = lane >= 16;
    if (tileBase >= npts) return;

    __builtin_prefetch(&x[tileBase], 0, 3);    // global_prefetch_b8

    int px = tileBase + laneN;
    if (px >= npts) px = npts - 1;
    const float xv = x[px];                    // point replicated in both lane halves

    v8f acc = {};                              // 16x16 f32 D: cols 0..2 = class sums

    // ---- single-site DFS state machine: "current node" + LDS stack ----
    int   sp    = 0;
    int   curId = 0, curD = 0;
    float curS  = 0.0f, curE = 1.0f, curW = 1.0f;
    bool  have  = true;

    while (have || sp > 0) {
        if (have) {
            have = false;
            float split = rflf(Lsplit[curId]);
            float seg   = curE - curS;
            float mid   = curS + split * seg;
            float t     = sigm((xv - mid) * (200.0f / seg));   // 20/(0.1*seg)
            if (curD == maxd - 1) {
                // Leaf parent: 4 leaf weights -> matrix pipe.
                // A (16x4 f32): lanes 0-15 K=0,1 (left leaves); 16-31 K=2,3 (right).
                int o = curId * 4;
                float blend = hi ? t : (1.0f - t);
                float q0 = hi ? Lp[o + 2] : Lp[o + 0];
                float q1 = hi ? Lp[o + 3] : Lp[o + 1];
                v2f A; A[0] = curW * blend * q0; A[1] = curW * blend * q1;
                // B (4x16 f32) one-hot by class: VGPR0 rows K=0(lo)/K=1(hi),
                // VGPR1 rows K=2(lo)/K=3(hi).
                int k0 = hi ? Lcls[o + 1] : Lcls[o + 0];
                int k1 = hi ? Lcls[o + 3] : Lcls[o + 2];
                v2f Bv; Bv[0] = (laneN == k0) ? 1.0f : 0.0f;
                        Bv[1] = (laneN == k1) ? 1.0f : 0.0f;
                acc = __builtin_amdgcn_wmma_f32_16x16x4_f32(
                    false, A, false, Bv, (short)0, acc, false, false);
            } else {
                if (lane == 0) {
                    Sid[wib][sp] = curId; Sst[wib][sp] = curS;
                    Sen[wib][sp] = curE;  Scc[wib][sp] = 0;
                }
                Sw[wib][sp][lane] = curW;
                St[wib][sp][lane] = t;
                ++sp;
            }
        } else {
            int d = sp - 1;
            int c = rfli(Scc[wib][d]);
            if (c >= 4) { --sp; continue; }
            if (lane == 0) Scc[wib][d] = c + 1;
            int id = rfli(Sid[wib][d]);
            float cp = rflf(Lp[id * 4 + c]);
            if (cp <= 0.0f) continue;          // uniform prune of zero-prob subtree
            float s  = rflf(Sst[wib][d]);
            float e  = rflf(Sen[wib][d]);
            float mid = s + rflf(Lsplit[id]) * (e - s);
            float w = Sw[wib][d][lane];
            float t = St[wib][d][lane];
            float blend = (c >= 2) ? t : (1.0f - t);
            curW  = w * blend * cp;
            curId = rfli(Lcid[id * 4 + c]);
            curS  = (c >= 2) ? mid : s;
            curE  = (c >= 2) ? e : mid;
            curD  = sp;                        // child depth = parent depth + 1
            have  = true;
        }
    }

    // ---- D layout: element (m,n) lives in VGPR m%8, lane n + 16*(m/8) ----
    const int hbase = hi ? 16 : 0;
    #pragma unroll
    for (int r = 0; r < 8; ++r) {
        float v  = acc[r];
        float c0 = __shfl(v, hbase + 0, 32);
        float c1 = __shfl(v, hbase + 1, 32);
        float c2 = __shfl(v, hbase + 2, 32);
        float total = c0 + c1 + c2;
        int m = (hi ? 8 : 0) + r;
        if (laneN < NUMV && (tileBase + m) < npts) {
            out[(tileBase + m) * NUMV + laneN] = 20.0f * v - 10.0f * total;
        }
    }
}

extern "C" void kernel_launch(void* const* d_in, const int* in_sizes, int n_in,
                              void* d_out, int out_size, void* d_ws, size_t ws_size,
                              hipStream_t stream) {
    const float*         x    = (const float*)d_in[0];
    const float*         spp  = (const float*)d_in[1];
    const float*         mats = (const float*)d_in[2];
    const unsigned char* lm   = (const unsigned char*)d_in[3];
    const unsigned char* rm   = (const unsigned char*)d_in[4];
    const int*           maxd = (const int*)d_in[5];
    float*               out  = (float*)d_out;

    int npts  = in_sizes[0];                   // 1024
    int waves = (npts + 15) / 16;              // one 16-point tile per wave
    int threads = waves * 32;
    int block = 256;                           // 8 wave32 per block
    int grid  = (threads + block - 1) / block;
    Fractal1D_kernel<<<grid, block, 0, stream>>>(x, spp, mats, lm, rm, maxd, out, npts);
}